// HandCraftedFeatures_32255204393793
// MI455X (gfx1250) — compile-verified
//
#include <hip/hip_runtime.h>
#include <hip/hip_bf16.h>
#include <math.h>

typedef __attribute__((ext_vector_type(2))) float v2f;
typedef __attribute__((ext_vector_type(8))) float v8f;

#define CROP 448
#define FULL 512
#define TOPO 32          // (512-448)/2
#define NR   56          // CROP/8 cells per side
#define NB   55          // NR - 2 + 1
#define ORI  9
#define NIMG 48

// ---------------------------------------------------------------------------
// Pass 1: one workgroup per (cell-row r, image b). 448 threads (14 waves).
//  A) stage sqrt(gray) rows [8r-1 .. 8r+8] into LDS (10 rows x 448 cols)
//  B) each thread = one pixel column: 8 rows of grad/mag/bin -> 9-bin column
//     histogram (registers), scaled by 1/64, stored to LDS (padded to 16)
//  C) waves 0..3: fold 8 columns -> 1 cell with chained V_WMMA_F32_16X16X4_F32
//     (A = ones indicator, exact f32), write hist[b][r][c][o] to workspace
// ---------------------------------------------------------------------------
__global__ __launch_bounds__(448) void hog_cells_kernel(
    const float* __restrict__ x, const float* __restrict__ coeffs,
    float* __restrict__ hist)
{
    __shared__ float lds_g[10][CROP];
    __shared__ float colhist[CROP][16];   // bins 9..15 zero-padded

    const int r   = blockIdx.x;           // cell row 0..55
    const int b   = blockIdx.y;           // image 0..47
    const int col = threadIdx.x;          // pixel column 0..447

    const float c0 = coeffs[0], c1 = coeffs[1], c2 = coeffs[2];

    // --- A: sqrt-gray rows into LDS -------------------------------------
    const size_t imgBase = (size_t)b * 3u * FULL * FULL;
    #pragma unroll
    for (int i = 0; i < 10; ++i) {
        int gy = 8 * r - 1 + i;           // row within crop
        float val = 0.0f;
        if (gy >= 0 && gy < CROP) {
            const float* px = x + imgBase + (size_t)(TOPO + gy) * FULL + (TOPO + col);
            float gsum = c0 * px[0]
                       + c1 * px[(size_t)FULL * FULL]
                       + c2 * px[2 * (size_t)FULL * FULL];
            val = sqrtf(gsum);
        }
        lds_g[i][col] = val;
    }
    __syncthreads();

    // --- B: per-column 9-bin histogram over the 8 strip rows -------------
    float acc[ORI];
    #pragma unroll
    for (int o = 0; o < ORI; ++o) acc[o] = 0.0f;

    #pragma unroll
    for (int y = 0; y < 8; ++y) {
        int gy = 8 * r + y;
        float gr = (gy == 0 || gy == CROP - 1)
                 ? 0.0f : (lds_g[y + 2][col] - lds_g[y][col]);
        float gc = (col == 0 || col == CROP - 1)
                 ? 0.0f : (lds_g[y + 1][col + 1] - lds_g[y + 1][col - 1]);
        float mag = sqrtf(gr * gr + gc * gc);
        float deg = atan2f(gr, gc) * 57.29577951308232f;  // degrees
        float m = fmodf(deg, 180.0f);
        if (m < 0.0f) m += 180.0f;                        // python-style mod
        int bin = (int)floorf(m * (1.0f / 20.0f));
        bin = bin < 0 ? 0 : (bin > 8 ? 8 : bin);
        #pragma unroll
        for (int o = 0; o < ORI; ++o) acc[o] += (bin == o) ? mag : 0.0f;
    }
    #pragma unroll
    for (int o = 0; o < 16; ++o)
        colhist[col][o] = (o < ORI) ? acc[o] * 0.015625f : 0.0f;  // /64 exact
    __syncthreads();

    // --- C: 8-column -> cell fold via chained f32 WMMA --------------------
    const int wave = threadIdx.x >> 5;
    const int lane = threadIdx.x & 31;
    if (wave < 4) {
        const int tile = wave;            // cells [16*tile, 16*tile+16)
        const int n  = lane & 15;         // bin column (valid < 9)
        const int kh = lane >> 4;         // lane-half -> K offset / M offset

        v8f c = {};                       // f32 accumulator, exact
        const int nslices = (tile == 3) ? 16 : 32;  // tile 3: cols 384..447
        for (int s = 0; s < nslices; ++s) {
            int base = tile * 128 + s * 4;            // 4 columns, one cell
            int mrow = (base >> 3) - tile * 16;       // cell within tile
            float av = ((lane & 15) == mrow) ? 1.0f : 0.0f;
            v2f a = { av, av };                       // K-invariant indicator
            v2f bb = { colhist[base + 2 * kh + 0][n],
                       colhist[base + 2 * kh + 1][n] };
            c = __builtin_amdgcn_wmma_f32_16x16x4_f32(
                    false, a, false, bb, (short)0, c, false, false);
        }

        if (n < ORI) {
            #pragma unroll
            for (int j = 0; j < 8; ++j) {
                int cell = tile * 16 + kh * 8 + j;    // C/D: VGPR j -> M=j(+8)
                if (cell < NR)
                    hist[(((size_t)b * NR + r) * NR + cell) * ORI + n] = c[j];
            }
        }
    }
}

// ---------------------------------------------------------------------------
// Pass 2: L2-Hys block normalization. One thread per (b, i, j) output block.
// ---------------------------------------------------------------------------
__global__ void hog_norm_kernel(const float* __restrict__ hist,
                                float* __restrict__ out)
{
    int idx = blockIdx.x * blockDim.x + threadIdx.x;
    const int total = NIMG * NB * NB;
    if (idx >= total) return;
    int j = idx % NB;
    int t = idx / NB;
    int i = t % NB;
    int b = t / NB;

    float v[36];
    float ss = 0.0f;
    #pragma unroll
    for (int s = 0; s < 4; ++s) {
        int ci = i + (s >> 1), cj = j + (s & 1);
        const float* hp = hist + (((size_t)b * NR + ci) * NR + cj) * ORI;
        #pragma unroll
        for (int o = 0; o < ORI; ++o) {
            float hv = hp[o];
            v[s * ORI + o] = hv;
            ss += hv * hv;
        }
    }
    float n1 = sqrtf(ss + 1e-10f);        // EPS*EPS
    float ss2 = 0.0f;
    #pragma unroll
    for (int k = 0; k < 36; ++k) {
        float u = v[k] / n1;
        u = u > 0.2f ? 0.2f : u;          // Hys clip
        v[k] = u;
        ss2 += u * u;
    }
    float n2 = sqrtf(ss2 + 1e-10f);
    float* op = out + (size_t)idx * 36;
    #pragma unroll
    for (int k = 0; k < 36; ++k) op[k] = v[k] / n2;
}

// ---------------------------------------------------------------------------
extern "C" void kernel_launch(void* const* d_in, const int* in_sizes, int n_in,
                              void* d_out, int out_size, void* d_ws, size_t ws_size,
                              hipStream_t stream) {
    (void)in_sizes; (void)n_in; (void)out_size; (void)ws_size;
    const float* x      = (const float*)d_in[0];   // (48,3,512,512) f32
    const float* coeffs = (const float*)d_in[1];   // (3,) f32
    float* hist = (float*)d_ws;                    // 48*56*56*9 f32 = 5.4 MB
    float* out  = (float*)d_out;                   // 48*55*55*36 f32

    dim3 grid1(NR, NIMG);
    hog_cells_kernel<<<grid1, CROP, 0, stream>>>(x, coeffs, hist);

    int total = NIMG * NB * NB;
    hog_norm_kernel<<<(total + 255) / 256, 256, 0, stream>>>(hist, out);
}